// GCN_52261162058422
// MI455X (gfx1250) — compile-verified
//
#include <hip/hip_runtime.h>
#include <hip/hip_bf16.h>

typedef __attribute__((ext_vector_type(2))) float v2f;
typedef __attribute__((ext_vector_type(8))) float v8f;

#define H 8
#define CDIM 8

// ---------------- zero helper ----------------
__global__ void k_zero(float* __restrict__ p, size_t n) {
    size_t i = (size_t)blockIdx.x * blockDim.x + threadIdx.x;
    size_t stride = (size_t)gridDim.x * blockDim.x;
    for (; i < n; i += stride) p[i] = 0.0f;
}

// ---------------- degree accumulation (4 edges/thread, b128 index loads) ---
__global__ void k_degrees(const int* __restrict__ src, const int* __restrict__ dst,
                          float* __restrict__ deg_in, float* __restrict__ deg_out, int E) {
    int base = (blockIdx.x * blockDim.x + threadIdx.x) * 4;
    if (base >= E) return;
    if (base + 4 <= E) {
        int4 s = *(const int4*)(src + base);
        int4 d = *(const int4*)(dst + base);
        atomicAdd(deg_in + d.x, 1.0f); atomicAdd(deg_in + d.y, 1.0f);
        atomicAdd(deg_in + d.z, 1.0f); atomicAdd(deg_in + d.w, 1.0f);
        atomicAdd(deg_out + s.x, 1.0f); atomicAdd(deg_out + s.y, 1.0f);
        atomicAdd(deg_out + s.z, 1.0f); atomicAdd(deg_out + s.w, 1.0f);
    } else {
        for (int e = base; e < E; ++e) {
            atomicAdd(deg_in + dst[e], 1.0f);
            atomicAdd(deg_out + src[e], 1.0f);
        }
    }
}

// ---------------- per-node prep: inv sqrt degrees, x0 = deg_in * inv_out ---
__global__ void k_prep(float* __restrict__ deg_in /*becomes x0*/,
                       const float* __restrict__ deg_out,
                       float* __restrict__ inv_in, float* __restrict__ inv_out, int N) {
    int n = blockIdx.x * blockDim.x + threadIdx.x;
    if (n >= N) return;
    float din  = deg_in[n];
    float dout = deg_out[n];
    float iin  = 1.0f / sqrtf(fmaxf(din,  1.0f));
    float iout = 1.0f / sqrtf(fmaxf(dout, 1.0f));
    inv_in[n]  = iin;
    inv_out[n] = iout;
    deg_in[n]  = din * iout;   // x0, overwrites deg_in (same-slot RMW, safe)
}

// ---------------- layer-1 edge aggregation (scalar feature, 4/thread) -----
__global__ void k_agg1(const int* __restrict__ src, const int* __restrict__ dst,
                       const float* __restrict__ x0, float* __restrict__ agg1, int E) {
    int base = (blockIdx.x * blockDim.x + threadIdx.x) * 4;
    if (base >= E) return;
    if (base + 4 <= E) {
        int4 s = *(const int4*)(src + base);
        int4 d = *(const int4*)(dst + base);
        float vx = x0[s.x], vy = x0[s.y], vz = x0[s.z], vw = x0[s.w];
        atomicAdd(agg1 + d.x, vx); atomicAdd(agg1 + d.y, vy);
        atomicAdd(agg1 + d.z, vz); atomicAdd(agg1 + d.w, vw);
    } else {
        for (int e = base; e < E; ++e) atomicAdd(agg1 + dst[e], x0[src[e]]);
    }
}

// ---------------- layer-1 node update: x1 = relu(s*W1) * inv_out ----------
__global__ void k_layer1(const float* __restrict__ agg1, const float* __restrict__ inv_in,
                         const float* __restrict__ inv_out, const float* __restrict__ W1,
                         float* __restrict__ x1, int N) {
    int n = blockIdx.x * blockDim.x + threadIdx.x;
    if (n >= N) return;
    float s = agg1[n] * inv_in[n];
    float io = inv_out[n];
    #pragma unroll
    for (int j = 0; j < H; ++j) {
        float v = s * W1[j];
        x1[(size_t)n * H + j] = (v > 0.0f ? v : 0.0f) * io;
    }
}

// ---------------- layer-2 edge aggregation (8 features, 4 edges/thread) ---
__global__ void k_agg2(const int* __restrict__ src, const int* __restrict__ dst,
                       const float* __restrict__ x1, float* __restrict__ agg2, int E) {
    int base = (blockIdx.x * blockDim.x + threadIdx.x) * 4;
    if (base >= E) return;
    if (base + 4 <= E) {
        int4 s4 = *(const int4*)(src + base);
        int4 d4 = *(const int4*)(dst + base);
        int sarr[4] = {s4.x, s4.y, s4.z, s4.w};
        int darr[4] = {d4.x, d4.y, d4.z, d4.w};
        #pragma unroll
        for (int i = 0; i < 4; ++i) {
            const float4* xr = (const float4*)(x1 + (size_t)sarr[i] * H);  // 2x b128 gather (L2)
            float4 lo = xr[0];
            float4 hi = xr[1];
            float* b = agg2 + (size_t)darr[i] * H;
            atomicAdd(b + 0, lo.x); atomicAdd(b + 1, lo.y);
            atomicAdd(b + 2, lo.z); atomicAdd(b + 3, lo.w);
            atomicAdd(b + 4, hi.x); atomicAdd(b + 5, hi.y);
            atomicAdd(b + 6, hi.z); atomicAdd(b + 7, hi.w);
        }
    } else {
        for (int e = base; e < E; ++e) {
            const float* xr = x1 + (size_t)src[e] * H;
            float* b = agg2 + (size_t)dst[e] * H;
            #pragma unroll
            for (int j = 0; j < H; ++j) atomicAdd(b + j, xr[j]);
        }
    }
}

// ---------------- layer-2 node update via fp32 WMMA -----------------------
// One wave per 16-node tile: D(16x16) = A(16x8 nodes) x B(8x16 = W2 padded),
// two chained V_WMMA_F32_16X16X4_F32. Buffers are tile-padded: no store guards.
__global__ void k_layer2_wmma(const float* __restrict__ agg2, const float* __restrict__ inv_in,
                              const float* __restrict__ W2, float* __restrict__ h2,
                              int N, int numTiles) {
    int wave = blockIdx.x * (blockDim.x >> 5) + (threadIdx.x >> 5);
    if (wave >= numTiles) return;            // wave-uniform: EXEC all-1s at the WMMAs
    int lane = threadIdx.x & 31;
    int half = lane >> 4;
    int l    = lane & 15;
    int n0   = wave * 16;

    // A fragment: row m=l of the tile (32-bit A layout: lanes 0-15 K={0,1}, 16-31 K={2,3})
    int node = n0 + l;
    int nclamp = node < N ? node : (N - 1);
    float iin = (node < N) ? inv_in[nclamp] : 0.0f;
    const float* arow = agg2 + (size_t)nclamp * H;
    int kb0 = half * 2;
    int kb1 = 4 + half * 2;
    v2f a0, a1;
    a0.x = arow[kb0 + 0] * iin;  a0.y = arow[kb0 + 1] * iin;
    a1.x = arow[kb1 + 0] * iin;  a1.y = arow[kb1 + 1] * iin;

    // B fragment: W2 (8x8 row-major) zero-padded to 4x16 per K-chunk
    v2f b0; b0.x = 0.0f; b0.y = 0.0f;
    v2f b1; b1.x = 0.0f; b1.y = 0.0f;
    if (l < H) {
        b0.x = W2[(kb0 + 0) * H + l];  b0.y = W2[(kb0 + 1) * H + l];
        b1.x = W2[(kb1 + 0) * H + l];  b1.y = W2[(kb1 + 1) * H + l];
    }

    v8f acc = {};
    acc = __builtin_amdgcn_wmma_f32_16x16x4_f32(false, a0, false, b0, (short)0, acc, false, false);
    acc = __builtin_amdgcn_wmma_f32_16x16x4_f32(false, a1, false, b1, (short)0, acc, false, false);

    // D: lane l = feature column, VGPR v = node (half*8 + v). h2 is tile-padded:
    // unconditional strided stores, single predicated region.
    if (l < H) {
        float* out = h2 + (size_t)(n0 + half * 8) * H + l;
        #pragma unroll
        for (int v = 0; v < 8; ++v) {
            float r = acc[v];
            out[(size_t)v * H] = r > 0.0f ? r : 0.0f;   // ReLU
        }
    }
}

// ---------------- graph pooling: sorted ids -> wave-level reduction -------
__global__ void k_pool(const float* __restrict__ h2, const int* __restrict__ graph_ids,
                       float* __restrict__ hg, float* __restrict__ counts, int N) {
    int n = blockIdx.x * blockDim.x + threadIdx.x;
    int n0 = n & ~31;               // first node of this wave's range
    if (n0 >= N) return;
    int lane = threadIdx.x & 31;
    bool valid = n < N;
    int nlast = n0 + 31 < N ? n0 + 31 : N - 1;

    // graph_ids is sorted: wave range is single-graph iff endpoints match
    int gfirst = graph_ids[n0];
    int glast  = graph_ids[nlast];

    float h[H];
    #pragma unroll
    for (int j = 0; j < H; ++j) h[j] = valid ? h2[(size_t)n * H + j] : 0.0f;

    if (gfirst == glast) {
        // wave-uniform graph: butterfly reduction across 32 lanes
        #pragma unroll
        for (int j = 0; j < H; ++j) {
            float v = h[j];
            #pragma unroll
            for (int off = 16; off > 0; off >>= 1)
                v += __shfl_xor(v, off, 32);
            h[j] = v;
        }
        if (lane == 0) {
            int nvalid = (N - n0) < 32 ? (N - n0) : 32;
            atomicAdd(counts + gfirst, (float)nvalid);
            float* dstp = hg + (size_t)gfirst * H;
            #pragma unroll
            for (int j = 0; j < H; ++j) atomicAdd(dstp + j, h[j]);
        }
    } else {
        if (valid) {
            int g = graph_ids[n];
            atomicAdd(counts + g, 1.0f);
            float* dstp = hg + (size_t)g * H;
            #pragma unroll
            for (int j = 0; j < H; ++j) atomicAdd(dstp + j, h[j]);
        }
    }
}

// ---------------- final: out = (hg / max(count,1)) @ Wlast ----------------
__global__ void k_final(const float* __restrict__ hg, const float* __restrict__ counts,
                        const float* __restrict__ Wlast, float* __restrict__ out, int G) {
    int g = blockIdx.x * blockDim.x + threadIdx.x;
    if (g >= G) return;
    float inv = 1.0f / fmaxf(counts[g], 1.0f);
    float h[H];
    #pragma unroll
    for (int k = 0; k < H; ++k) h[k] = hg[(size_t)g * H + k] * inv;
    #pragma unroll
    for (int c = 0; c < CDIM; ++c) {
        float s = 0.0f;
        #pragma unroll
        for (int k = 0; k < H; ++k) s += h[k] * Wlast[k * CDIM + c];
        out[(size_t)g * CDIM + c] = s;
    }
}

extern "C" void kernel_launch(void* const* d_in, const int* in_sizes, int n_in,
                              void* d_out, int out_size, void* d_ws, size_t ws_size,
                              hipStream_t stream) {
    const float* W1    = (const float*)d_in[0];   // 1 x H
    const float* W2    = (const float*)d_in[1];   // H x H
    const float* Wlast = (const float*)d_in[2];   // H x C
    const int*   src   = (const int*)d_in[3];     // E
    const int*   dst   = (const int*)d_in[4];     // E
    const int*   gids  = (const int*)d_in[5];     // N

    const int E = in_sizes[3];
    const int N = in_sizes[5];
    const int G = out_size / CDIM;
    const int numTiles = (N + 15) / 16;
    const size_t Np = (size_t)numTiles * 16;      // tile-padded node count

    // ---- workspace layout (floats, 16B-aligned sub-buffers) ----
    float* ws = (float*)d_ws;
    size_t off = 0;
    auto alloc = [&](size_t n) { float* p = ws + off; off += (n + 3) & ~(size_t)3; return p; };
    float* deg_in  = alloc(N);            // becomes x0 after k_prep
    float* deg_out = alloc(N);
    float* inv_in  = alloc(N);
    float* inv_out = alloc(N);
    float* agg1    = alloc(N);
    float* x1      = alloc(Np * H);       // tile-padded; becomes h2 after layer 2
    float* agg2    = alloc(Np * H);       // tile-padded
    float* hg      = alloc((size_t)G * H);
    float* cnts    = alloc(G);
    float* x0 = deg_in;
    float* h2 = x1;
    (void)ws_size; (void)n_in;

    const int T = 256;
    const int ZB = 2048;

    // zero accumulators each call (harness does not re-poison between replays)
    k_zero<<<ZB, T, 0, stream>>>(deg_in,  N);
    k_zero<<<ZB, T, 0, stream>>>(deg_out, N);
    k_zero<<<ZB, T, 0, stream>>>(agg1,    N);
    k_zero<<<ZB, T, 0, stream>>>(agg2,    Np * H);   // includes tile pad
    k_zero<<<ZB, T, 0, stream>>>(hg,     (size_t)G * H);
    k_zero<<<ZB, T, 0, stream>>>(cnts,    G);

    int eb4 = (E + 4 * T - 1) / (4 * T);
    int nb  = (N + T - 1) / T;
    int gb  = (G + T - 1) / T;

    k_degrees<<<eb4, T, 0, stream>>>(src, dst, deg_in, deg_out, E);
    k_prep   <<<nb,  T, 0, stream>>>(deg_in, deg_out, inv_in, inv_out, N);
    k_agg1   <<<eb4, T, 0, stream>>>(src, dst, x0, agg1, E);
    k_layer1 <<<nb,  T, 0, stream>>>(agg1, inv_in, inv_out, W1, x1, N);
    k_agg2   <<<eb4, T, 0, stream>>>(src, dst, x1, agg2, E);

    int wavesPerBlock = T / 32;
    int tb = (numTiles + wavesPerBlock - 1) / wavesPerBlock;
    k_layer2_wmma<<<tb, T, 0, stream>>>(agg2, inv_in, W2, h2, N, numTiles);

    k_pool <<<nb, T, 0, stream>>>(h2, gids, hg, cnts, N);
    k_final<<<gb, T, 0, stream>>>(hg, cnts, Wlast, (float*)d_out, G);
}